// ModelNew_50388556316983
// MI455X (gfx1250) — compile-verified
//
#include <hip/hip_runtime.h>

typedef __attribute__((ext_vector_type(2))) float v2f;
typedef __attribute__((ext_vector_type(8))) float v8f;

#define OC      16
#define IC      3
#define D_IN    128
#define D_OUT   127
#define KTOT    24          // IC * 2*2*2
#define WTILES  8           // ceil(127/16)
#define CH_STRIDE 2048383   // 127*127*127

// Workspace layout (floats): [0..383] fwT[k=24][oc=16], [384..399] fb[16]
__global__ void ModelNew_prep(const float* __restrict__ w,   // (3,16,3,3,3)
                              const float* __restrict__ cb,  // (16,)
                              const float* __restrict__ eb,  // (16,)
                              float* __restrict__ ws) {
  int t = threadIdx.x;
  if (t < KTOT * OC) {
    int oc = t & 15;
    int k  = t >> 4;                      // k = ic*8 + kd*4 + kh*2 + kw
    int ic = k >> 3;
    int kd = (k >> 2) & 1, kh = (k >> 1) & 1, kw = k & 1;
    // fold: tap index 0 -> sum taps {1,2}; tap index 1 -> tap {0}
    int as[2] = { kd ? 0 : 1, kd ? 0 : 2 }; int na = kd ? 1 : 2;
    int bs[2] = { kh ? 0 : 1, kh ? 0 : 2 }; int nb = kh ? 1 : 2;
    int cs[2] = { kw ? 0 : 1, kw ? 0 : 2 }; int nc = kw ? 1 : 2;
    const float* wb = w + (ic * OC + oc) * 27;
    float s = 0.f;
    for (int i = 0; i < na; ++i)
      for (int j = 0; j < nb; ++j)
        for (int l = 0; l < nc; ++l)
          s += wb[as[i] * 9 + bs[j] * 3 + cs[l]];
    ws[k * OC + oc] = 0.0625f * s;        // SCALE1*SCALE2*0.125
  }
  if (t < OC) ws[KTOT * OC + t] = cb[t] * 0.5f + eb[t];
}

// One wave computes a 16(oc) x 16(ow) output tile for a given (n, od, oh, ow0)
// via 6 chained V_WMMA_F32_16X16X4_F32 (K = 24 in tiles of 4).
__global__ __launch_bounds__(256) void ModelNew_conv(
    const float* __restrict__ x,        // (4,3,128,128,128)
    const float* __restrict__ ws,
    float* __restrict__ out) {          // (4,16,127,127,127)
  const int lane = threadIdx.x & 31;
  const int hs   = lane >> 4;           // half-wave select (K pair offset)
  const int ncol = lane & 15;           // M for A, N for B/C/D
  const int wave = threadIdx.x >> 5;

  int tile = blockIdx.x * 8 + wave;     // grid sized for exact fit
  int wt   = tile & (WTILES - 1);
  int rest = tile >> 3;
  int oh   = rest % D_OUT;  rest /= D_OUT;
  int od   = rest % D_OUT;
  int n    = rest / D_OUT;
  int ow0  = wt * 16;

  const float* fwT = ws;
  const float* fb  = ws + KTOT * OC;

  // ---- A fragments (weights): lane holds M=ncol, K = {4t+2hs, 4t+2hs+1}
  v2f a[6];
#pragma unroll
  for (int t = 0; t < 6; ++t) {
    int khi = 2 * t + hs;               // = k>>1 for this lane's K pair
    a[t].x = fwT[khi * 32 + ncol];      // k = 2*khi   (kw=0)
    a[t].y = fwT[khi * 32 + 16 + ncol]; // k = 2*khi+1 (kw=1)
  }

  // ---- C init with fused bias: VGPR r -> oc = r + 8*hs
  v8f c;
#pragma unroll
  for (int r = 0; r < 8; ++r) c[r] = fb[hs * 8 + r];

  // ---- B fragments (input patches): lane holds N=ncol, same K pair as A
  int iw0 = ow0 + ncol;     if (iw0 > D_IN - 1) iw0 = D_IN - 1;  // clamp (masked col)
  int iw1 = ow0 + ncol + 1; if (iw1 > D_IN - 1) iw1 = D_IN - 1;
  v2f b[6];
#pragma unroll
  for (int t = 0; t < 6; ++t) {
    int khi = 2 * t + hs;
    int ic = khi >> 2, kd = (khi >> 1) & 1, kh = khi & 1;
    const float* p = x + (((size_t)(n * IC + ic) * D_IN + (od + kd)) * D_IN
                          + (oh + kh)) * D_IN;
    b[t].x = p[iw0];                    // kw = 0
    b[t].y = p[iw1];                    // kw = 1
  }

  // ---- 6x chained fp32 WMMA: D = A(16x4) * B(4x16) + C
#pragma unroll
  for (int t = 0; t < 6; ++t)
    c = __builtin_amdgcn_wmma_f32_16x16x4_f32(
        /*neg_a=*/false, a[t], /*neg_b=*/false, b[t],
        /*c_mod=*/(short)0, c, /*reuse_a=*/false, /*reuse_b=*/false);

  // ---- store: VGPR r, lanes 0-15 -> oc=r, lanes 16-31 -> oc=r+8
  int ow = ow0 + ncol;
  if (ow < D_OUT) {
    size_t base = (((size_t)n * OC + hs * 8) * D_OUT + od) * (size_t)(D_OUT * D_OUT)
                + (size_t)oh * D_OUT + (size_t)ow;
#pragma unroll
    for (int r = 0; r < 8; ++r)
      out[base + (size_t)r * CH_STRIDE] = c[r];
  }
}

extern "C" void kernel_launch(void* const* d_in, const int* in_sizes, int n_in,
                              void* d_out, int out_size, void* d_ws, size_t ws_size,
                              hipStream_t stream) {
  const float* x  = (const float*)d_in[0];  // (4,3,128,128,128)
  const float* cw = (const float*)d_in[1];  // (3,16,3,3,3)
  const float* cb = (const float*)d_in[2];  // (16,)
  const float* eb = (const float*)d_in[3];  // (16,1,1,1)
  float* out = (float*)d_out;
  float* ws  = (float*)d_ws;

  ModelNew_prep<<<1, 384, 0, stream>>>(cw, cb, eb, ws);

  const int n_tiles = 4 * D_OUT * D_OUT * WTILES;   // 516128, multiple of 8
  ModelNew_conv<<<n_tiles / 8, 256, 0, stream>>>(x, ws, out);
}